// LocalAttention_28518582845970
// MI455X (gfx1250) — compile-verified
//
#include <hip/hip_runtime.h>

typedef __attribute__((ext_vector_type(2))) float v2f;
typedef __attribute__((ext_vector_type(8))) float v8f;

#define EMBED 1024
#define IMGW  256
#define WIN   64
#define NW    129   // window length along the row

// ---------------------------------------------------------------------------
// Kernel 1: Q/K/V projection of the 129 needed pixels (image row px,
// cols py-64 .. py+64), Y = X * W^T + b, via V_WMMA_F32_16X16X4_F32.
// One wave computes one 16(M) x 16(N) f32 tile; K-loop over EMBED in steps
// of 4 with two accumulators to break the WMMA->WMMA RAW chain.
// M padded to 144 rows; padded reads are in-bounds (image is 256 wide) and
// padded rows are never stored.
// ---------------------------------------------------------------------------
__global__ __launch_bounds__(32)
void qkv_wmma_kernel(const float* __restrict__ matrix,
                     const float* __restrict__ Wq, const float* __restrict__ bq,
                     const float* __restrict__ Wk, const float* __restrict__ bk,
                     const float* __restrict__ Wv, const float* __restrict__ bv,
                     const int*   __restrict__ ppx, const int* __restrict__ ppy,
                     float* __restrict__ YQ, float* __restrict__ YK,
                     float* __restrict__ YV)
{
    const int lane = threadIdx.x;            // 0..31, wave32
    const int n0   = blockIdx.x * 16;        // feature tile
    const int m0   = blockIdx.y * 16;        // pixel tile (0..143)

    const float* W; const float* bias; float* Y;
    if (blockIdx.z == 0)      { W = Wq; bias = bq; Y = YQ; }
    else if (blockIdx.z == 1) { W = Wk; bias = bk; Y = YK; }
    else                      { W = Wv; bias = bv; Y = YV; }

    const int px = ppx[0], py = ppy[0];
    const float* X = matrix + ((size_t)px * IMGW + (size_t)(py - WIN)) * EMBED;

    // A fragment (16x4 f32): lanes 0-15 -> K={0,1}, lanes 16-31 -> K={2,3}
    // B fragment (4x16 f32): same K split; N = lane&15. B[e][n] = W[n][e].
    const int kb = (lane >> 4) * 2;          // 0 or 2
    const float* Xrow = X + (size_t)(m0 + (lane & 15)) * EMBED + kb;
    const float* Wrow = W + (size_t)(n0 + (lane & 15)) * EMBED + kb;

    v8f acc0 = {}; v8f acc1 = {};
    #pragma unroll 4
    for (int k = 0; k < EMBED; k += 8) {
        v2f a0 = *(const v2f*)(Xrow + k);
        v2f b0 = *(const v2f*)(Wrow + k);
        v2f a1 = *(const v2f*)(Xrow + k + 4);
        v2f b1 = *(const v2f*)(Wrow + k + 4);
        acc0 = __builtin_amdgcn_wmma_f32_16x16x4_f32(false, a0, false, b0,
                                                     (short)0, acc0, false, false);
        acc1 = __builtin_amdgcn_wmma_f32_16x16x4_f32(false, a1, false, b1,
                                                     (short)0, acc1, false, false);
    }
    v8f acc = acc0 + acc1;

    // C/D layout: VGPR i -> M = m0 + i + 8*(lane>=16), N = n0 + (lane&15)
    const int   n  = n0 + (lane & 15);
    const float bn = bias[n];
    const int   mh = m0 + ((lane >> 4) * 8);
    #pragma unroll
    for (int i = 0; i < 8; ++i) {
        int m = mh + i;
        if (m < NW) Y[(size_t)m * EMBED + n] = acc[i] + bn;
    }
}

// ---------------------------------------------------------------------------
// Kernel 2: scores[k] = (q . K_k) / sqrt(EMBED), one wave per k.
// ---------------------------------------------------------------------------
__global__ __launch_bounds__(128)
void scores_kernel(const float* __restrict__ YQ, const float* __restrict__ YK,
                   float* __restrict__ scores)
{
    const int wave = threadIdx.x >> 5;
    const int lane = threadIdx.x & 31;
    const int k = blockIdx.x * 4 + wave;
    if (k >= NW) return;
    const float* q  = YQ + (size_t)WIN * EMBED;   // q is window index 64
    const float* kr = YK + (size_t)k * EMBED;
    float s = 0.f;
    for (int e = lane; e < EMBED; e += 32) s += q[e] * kr[e];
    #pragma unroll
    for (int off = 16; off; off >>= 1) s += __shfl_down(s, off, 32);
    if (lane == 0) scores[k] = s * 0.03125f;      // 1/sqrt(1024)
}

// ---------------------------------------------------------------------------
// Kernel 3: softmax over 129 scores (single block).
// ---------------------------------------------------------------------------
__global__ __launch_bounds__(256)
void softmax_kernel(const float* __restrict__ scores, float* __restrict__ p)
{
    __shared__ float sh[256];
    const int t = threadIdx.x;
    float v = (t < NW) ? scores[t] : -3.0e38f;
    sh[t] = v; __syncthreads();
    for (int s = 128; s; s >>= 1) {
        if (t < s) sh[t] = fmaxf(sh[t], sh[t + s]);
        __syncthreads();
    }
    const float mx = sh[0]; __syncthreads();
    float e = (t < NW) ? __expf(v - mx) : 0.f;
    sh[t] = e; __syncthreads();
    for (int s = 128; s; s >>= 1) {
        if (t < s) sh[t] += sh[t + s];
        __syncthreads();
    }
    const float inv = 1.0f / sh[0];
    if (t < NW) p[t] = e * inv;
}

// ---------------------------------------------------------------------------
// Kernel 4: ctx[e] = sum_k p[k] * V_k[e]   (coalesced over e).
// ---------------------------------------------------------------------------
__global__ __launch_bounds__(128)
void ctx_kernel(const float* __restrict__ p, const float* __restrict__ YV,
                float* __restrict__ ctx)
{
    __shared__ float pp[NW];
    for (int i = threadIdx.x; i < NW; i += 128) pp[i] = p[i];
    __syncthreads();
    const int e = blockIdx.x * 128 + threadIdx.x;
    float acc = 0.f;
    for (int k = 0; k < NW; ++k) acc += pp[k] * YV[(size_t)k * EMBED + e];
    ctx[e] = acc;
}

// ---------------------------------------------------------------------------
// Kernel 5: out[f] = Wo[f] . ctx + bo[f], one wave per output feature.
// ---------------------------------------------------------------------------
__global__ __launch_bounds__(128)
void outproj_kernel(const float* __restrict__ Wo, const float* __restrict__ bo,
                    const float* __restrict__ ctx, float* __restrict__ out)
{
    const int wave = threadIdx.x >> 5;
    const int lane = threadIdx.x & 31;
    const int f = blockIdx.x * 4 + wave;
    const float* wr = Wo + (size_t)f * EMBED;
    float s = 0.f;
    for (int e = lane; e < EMBED; e += 32) s += wr[e] * ctx[e];
    #pragma unroll
    for (int off = 16; off; off >>= 1) s += __shfl_down(s, off, 32);
    if (lane == 0) out[f] = s + bo[f];
}

// ---------------------------------------------------------------------------
extern "C" void kernel_launch(void* const* d_in, const int* in_sizes, int n_in,
                              void* d_out, int out_size, void* d_ws, size_t ws_size,
                              hipStream_t stream)
{
    const float* matrix = (const float*)d_in[0];
    const float* Wq = (const float*)d_in[1];
    const float* bq = (const float*)d_in[2];
    const float* Wk = (const float*)d_in[3];
    const float* bk = (const float*)d_in[4];
    const float* Wv = (const float*)d_in[5];
    const float* bv = (const float*)d_in[6];
    const float* Wo = (const float*)d_in[7];
    const float* bo = (const float*)d_in[8];
    const int*  ppx = (const int*)d_in[9];
    const int*  ppy = (const int*)d_in[10];

    float* ws     = (float*)d_ws;
    float* YQ     = ws;                  // 129*1024
    float* YK     = ws + 132096;         // 129*1024
    float* YV     = ws + 264192;         // 129*1024
    float* scores = ws + 396288;         // 129
    float* p      = ws + 396544;         // 129
    float* ctx    = ws + 396800;         // 1024   (total ~1.6 MB)

    qkv_wmma_kernel<<<dim3(64, 9, 3), 32, 0, stream>>>(
        matrix, Wq, bq, Wk, bk, Wv, bv, ppx, ppy, YQ, YK, YV);
    scores_kernel <<<33, 128, 0, stream>>>(YQ, YK, scores);
    softmax_kernel<<<1, 256, 0, stream>>>(scores, p);
    ctx_kernel    <<<8, 128, 0, stream>>>(p, YV, ctx);
    outproj_kernel<<<256, 128, 0, stream>>>(Wo, bo, ctx, (float*)d_out);
}